// ResidualVectorQuantizer_77867757076523
// MI455X (gfx1250) — compile-verified
//
#include <hip/hip_runtime.h>
#include <math.h>

typedef __attribute__((ext_vector_type(2))) float v2f;
typedef __attribute__((ext_vector_type(8))) float v8f;

#define BETA_W      0.01f
#define B_ROWS      65536
#define D_DIM       128
#define K_CODES     256
#define L_LEVELS    4
#define WAVES_PB    8
#define ROWS_PW     16
#define ROWS_PB     (WAVES_PB * ROWS_PW)   // 128 rows per block

__global__ __launch_bounds__(256)
void rvq_gfx1250_kernel(const float* __restrict__ x,
                        const float* __restrict__ codebooks,
                        float* __restrict__ out) {
    __shared__ float cn_lds[K_CODES];                 // ||c||^2 for current level
    __shared__ int   idx_lds[WAVES_PB][ROWS_PW];      // chosen code per row

    const int tid  = threadIdx.x;
    const int wid  = tid >> 5;
    const int lane = tid & 31;
    const int half = lane >> 4;     // 0: K-cols {0,1}, 1: K-cols {2,3} of each chunk
    const int l16  = lane & 15;

    const int rowg0 = blockIdx.x * ROWS_PB + wid * ROWS_PW;  // first row of this wave
    const int myrow = rowg0 + l16;                           // row this lane carries

    // Output layout (flat concatenation, all float):
    float* out_idx   = out;                                                    // [B,L]
    float* out_prob  = out + (size_t)B_ROWS * L_LEVELS;                        // [B,K,L]
    float* out_quant = out_prob + (size_t)B_ROWS * K_CODES * L_LEVELS;         // [B,D]
    float* out_loss  = out_quant + (size_t)B_ROWS * D_DIM;                     // [B,L]

    // Residual A-fragment: 32 k-chunks x float2; element (2c+j) = r[myrow][4c+2h+j]
    float ra[64];
    {
        const float* xr = x + (size_t)myrow * D_DIM + 2 * half;
        #pragma unroll
        for (int c = 0; c < 32; ++c) {
            v2f v = *(const v2f*)(xr + 4 * c);
            ra[2 * c]     = v.x;
            ra[2 * c + 1] = v.y;
        }
    }

    for (int l = 0; l < L_LEVELS; ++l) {
        const float* cb = codebooks + (size_t)l * K_CODES * D_DIM;

        // ---- codebook squared norms -> LDS (one code per thread) ----
        __syncthreads();  // protect cn_lds / idx_lds reuse across levels
        {
            const float* cr = cb + (size_t)tid * D_DIM;
            float s = 0.f;
            #pragma unroll
            for (int c = 0; c < 32; ++c) {
                v2f a = *(const v2f*)(cr + 4 * c);
                v2f b = *(const v2f*)(cr + 4 * c + 2);
                s += a.x * a.x + a.y * a.y + b.x * b.x + b.y * b.y;
            }
            cn_lds[tid] = s;
        }
        __syncthreads();

        // B-fragment base for this lane: codebook row (nbase + l16), col 2h
        const float* cbl = cb + (size_t)l16 * D_DIM + 2 * half;

        // ---- pass 1: online softmax stats + argmin over 16 n-tiles ----
        float m[8], z[8], best[8];
        int   bidx[8];
        #pragma unroll
        for (int i = 0; i < 8; ++i) { m[i] = -3.0e38f; z[i] = 0.f; best[i] = -3.0e38f; bidx[i] = 0; }

        for (int t = 0; t < 16; ++t) {
            const int nbase = t * 16;
            const float* cbt = cbl + (size_t)nbase * D_DIM;
            v8f acc = {};
            #pragma unroll
            for (int c = 0; c < 32; ++c) {
                v2f a; a.x = ra[2 * c]; a.y = ra[2 * c + 1];
                v2f b = *(const v2f*)(cbt + 4 * c);
                acc = __builtin_amdgcn_wmma_f32_16x16x4_f32(
                        false, a, false, b, (short)0, acc, false, false);
            }
            const float cn  = cn_lds[nbase + l16];
            const int  kidx = nbase + l16;
            #pragma unroll
            for (int i = 0; i < 8; ++i) {
                float tv = 2.0f * acc[i] - cn;     // == -(d) up to per-row const
                if (tv > best[i]) { best[i] = tv; bidx[i] = kidx; }  // ascending k => first-occurrence argmin
                if (tv > m[i]) { z[i] = z[i] * __expf(m[i] - tv) + 1.0f; m[i] = tv; }
                else           { z[i] += __expf(tv - m[i]); }
            }
        }

        // butterfly reduction across the 16 lanes of each half (rows stay put)
        #pragma unroll
        for (int off = 1; off < 16; off <<= 1) {
            #pragma unroll
            for (int i = 0; i < 8; ++i) {
                float om = __shfl_xor(m[i],    off, 32);
                float oz = __shfl_xor(z[i],    off, 32);
                float ob = __shfl_xor(best[i], off, 32);
                int   oi = __shfl_xor(bidx[i], off, 32);
                float nm = fmaxf(m[i], om);
                z[i] = z[i] * __expf(m[i] - nm) + oz * __expf(om - nm);
                m[i] = nm;
                if (ob > best[i] || (ob == best[i] && oi < bidx[i])) { best[i] = ob; bidx[i] = oi; }
            }
        }

        // publish chosen indices (rows 0-7 from lane 0, rows 8-15 from lane 16)
        if (lane == 0) {
            #pragma unroll
            for (int i = 0; i < 8; ++i) idx_lds[wid][i] = bidx[i];
        }
        if (lane == 16) {
            #pragma unroll
            for (int i = 0; i < 8; ++i) idx_lds[wid][8 + i] = bidx[i];
        }
        __syncthreads();

        // ---- pass 2: recompute scores, write normalized probs ----
        float rz[8];
        #pragma unroll
        for (int i = 0; i < 8; ++i) rz[i] = 1.0f / z[i];

        for (int t = 0; t < 16; ++t) {
            const int nbase = t * 16;
            const float* cbt = cbl + (size_t)nbase * D_DIM;
            v8f acc = {};
            #pragma unroll
            for (int c = 0; c < 32; ++c) {
                v2f a; a.x = ra[2 * c]; a.y = ra[2 * c + 1];
                v2f b = *(const v2f*)(cbt + 4 * c);
                acc = __builtin_amdgcn_wmma_f32_16x16x4_f32(
                        false, a, false, b, (short)0, acc, false, false);
            }
            const float cn = cn_lds[nbase + l16];
            const int  k   = nbase + l16;
            #pragma unroll
            for (int i = 0; i < 8; ++i) {
                const int rowg = rowg0 + i + 8 * half;
                float tv = 2.0f * acc[i] - cn;
                float p  = __expf(tv - m[i]) * rz[i];
                out_prob[((size_t)rowg * K_CODES + k) * L_LEVELS + l] = p;
            }
        }

        // ---- gather chosen code, loss, residual update ----
        const int myidx = idx_lds[wid][l16];
        const float* qr = cb + (size_t)myidx * D_DIM + 2 * half;
        float sq = 0.f;
        #pragma unroll
        for (int c = 0; c < 32; ++c) {
            v2f q = *(const v2f*)(qr + 4 * c);
            float d0 = ra[2 * c]     - q.x;
            float d1 = ra[2 * c + 1] - q.y;
            sq += d0 * d0 + d1 * d1;
            ra[2 * c]     = d0;
            ra[2 * c + 1] = d1;
        }
        sq += __shfl_xor(sq, 16, 32);   // combine the two half-rows
        if (half == 0) {
            out_loss[(size_t)myrow * L_LEVELS + l] = (1.0f + BETA_W) * sq * (1.0f / (float)D_DIM);
            out_idx [(size_t)myrow * L_LEVELS + l] = (float)myidx;
        }
    }

    // quantized = x - final residual
    {
        const float* xr = x + (size_t)myrow * D_DIM + 2 * half;
        float* qo = out_quant + (size_t)myrow * D_DIM + 2 * half;
        #pragma unroll
        for (int c = 0; c < 32; ++c) {
            v2f v = *(const v2f*)(xr + 4 * c);
            v2f w;
            w.x = v.x - ra[2 * c];
            w.y = v.y - ra[2 * c + 1];
            *(v2f*)(qo + 4 * c) = w;
        }
    }
}

extern "C" void kernel_launch(void* const* d_in, const int* in_sizes, int n_in,
                              void* d_out, int out_size, void* d_ws, size_t ws_size,
                              hipStream_t stream) {
    (void)in_sizes; (void)n_in; (void)out_size; (void)d_ws; (void)ws_size;
    const float* x  = (const float*)d_in[0];
    const float* cb = (const float*)d_in[1];
    float* out = (float*)d_out;

    dim3 grid(B_ROWS / ROWS_PB);   // 512 blocks
    dim3 block(256);               // 8 waves, each owning 16 rows
    rvq_gfx1250_kernel<<<grid, block, 0, stream>>>(x, cb, out);
}